// NemotronHAttentionMixer_12360915878723
// MI455X (gfx1250) — compile-verified
//
#include <hip/hip_runtime.h>

typedef __attribute__((ext_vector_type(2))) float        v2f;
typedef __attribute__((ext_vector_type(4))) float        v4f;
typedef __attribute__((ext_vector_type(8))) float        v8f;
typedef __attribute__((ext_vector_type(4))) unsigned int v4u;
typedef __attribute__((ext_vector_type(8))) unsigned int v8u;

constexpr int B  = 32;
constexpr int S  = 8;
constexpr int H  = 4096;
constexpr int HQ = 32;
constexpr int HK = 8;
constexpr int D  = 128;
constexpr int L  = 4096;
constexpr int HALF = D / 2;
constexpr int G  = HQ / HK;   // 4

// ---------------------------------------------------------------------------
// TDM: issue a tensor_load_to_lds of a 2D tile (16 rows x 128 f32, row stride
// HK*D elements) into LDS at byte offset lds_off. Descriptor per CDNA5 ISA
// CDNA5 ISA. Wave-level op, EXEC ignored, TENSORcnt tracked.
// ---------------------------------------------------------------------------
__device__ __forceinline__ void tdm_load_tile_16x128(unsigned lds_off,
                                                     const float* src)
{
    unsigned long long ga = (unsigned long long)src;

    v4u g0;
    g0[0] = 1u;                                    // count=1 (valid), no gather
    g0[1] = lds_off;                               // LDS byte address
    g0[2] = (unsigned)(ga & 0xffffffffu);          // global_addr[31:0]
    g0[3] = ((unsigned)(ga >> 32) & 0x01ffffffu)   // global_addr[56:32]
            | 0x80000000u;                         // type=2 ("image") @127:126

    v8u g1;
    g1[0] = 2u << 16;                              // data_size=2 (4 bytes)
    g1[1] = (unsigned)D << 16;                     // tensor_dim0=128 @ bits 79:48
    g1[2] = ((unsigned)(L * HK) & 0xffffu) << 16;  // tensor_dim1 lo @ bits 111:80
    g1[3] = ((unsigned)(L * HK) >> 16)             // tensor_dim1 hi
            | ((unsigned)D << 16);                 // tile_dim0=128 @ bits 127:112
    g1[4] = 16u;                                   // tile_dim1=16 rows
    g1[5] = (unsigned)(HK * D);                    // tensor_dim0_stride=1024 elems
    g1[6] = 0u;
    g1[7] = 0u;

    asm volatile("tensor_load_to_lds %0, %1" :: "s"(g0), "s"(g1) : "memory");
}

// ---------------------------------------------------------------------------
// Generic GEMM: Y[M,N] = X[M,K] @ W[N,K]^T  using V_WMMA_F32_16X16X4_F32.
// 4 waves per block, each wave owns one 16x16 output tile (block tile 16x64).
// Each lane walks its row contiguously over k, so the clause'd b64 loads
// consume full cachelines (verified in round-1 asm).
// ---------------------------------------------------------------------------
__global__ __launch_bounds__(128)
void gemm_xwt_f32(const float* __restrict__ X, const float* __restrict__ W,
                  float* __restrict__ Y, int M, int N, int K)
{
    const int wave = threadIdx.x >> 5;
    const int lane = threadIdx.x & 31;
    const int hl   = lane >> 4;        // 0 or 1
    const int r    = lane & 15;

    const int m0 = blockIdx.y * 16;
    const int n0 = blockIdx.x * 64 + wave * 16;

    const float* xrow = X + (size_t)(m0 + r) * K + 2 * hl;
    const float* wrow = W + (size_t)(n0 + r) * K + 2 * hl;

    v8f acc = {};
    for (int k = 0; k < K; k += 4) {
        v2f a = *(const v2f*)(xrow + k);
        v2f b = *(const v2f*)(wrow + k);
        acc = __builtin_amdgcn_wmma_f32_16x16x4_f32(
            false, a, false, b, (short)0, acc, false, false);
    }

    const int mb = m0 + hl * 8;
#pragma unroll
    for (int i = 0; i < 8; ++i)
        Y[(size_t)(mb + i) * N + n0 + r] = acc[i];
}

// ---------------------------------------------------------------------------
// RoPE (in place) on q (B,S,HQ,D) and k (B,S,HK,D). cs: (B,S,2,HALF).
// ---------------------------------------------------------------------------
__global__ void rope_kernel(float* __restrict__ q, float* __restrict__ k,
                            const float* __restrict__ cs)
{
    int idx = blockIdx.x * blockDim.x + threadIdx.x;
    const int total = B * S * (HQ + HK) * HALF;
    if (idx >= total) return;

    int d   = idx % HALF;
    int row = idx / HALF;
    int h   = row % (HQ + HK);
    int bs  = row / (HQ + HK);

    float c = cs[(size_t)bs * 2 * HALF + d];
    float s = cs[(size_t)bs * 2 * HALF + HALF + d];

    float* base = (h < HQ)
        ? q + ((size_t)bs * HQ + h) * D
        : k + ((size_t)bs * HK + (h - HQ)) * D;

    float x1 = base[d];
    float x2 = base[d + HALF];
    base[d]        = x1 * c - x2 * s;
    base[d + HALF] = x2 * c + x1 * s;
}

// ---------------------------------------------------------------------------
// Stream past_key_value (1 GB) into the output. Source is read exactly once:
// non-temporal loads keep it from washing L2 (192 MB), which we want warm for
// the attention pass over the freshly written caches. Stores stay RT.
// ---------------------------------------------------------------------------
__global__ void cache_copy(const v4f* __restrict__ src,
                           v4f* __restrict__ dst, size_t n4)
{
    size_t i = (size_t)blockIdx.x * blockDim.x + threadIdx.x;
    size_t stride = (size_t)gridDim.x * blockDim.x;
    for (; i < n4; i += stride)
        dst[i] = __builtin_nontemporal_load(&src[i]);
}

// ---------------------------------------------------------------------------
// Scatter new (RoPE'd) K and V rows into caches at pos = start[b] + s.
// ---------------------------------------------------------------------------
__global__ void cache_scatter(const float* __restrict__ kn,
                              const float* __restrict__ vn,
                              const int* __restrict__ start,
                              float* __restrict__ kc, float* __restrict__ vc)
{
    int idx = blockIdx.x * blockDim.x + threadIdx.x;
    const int total = B * S * HK * D;
    if (idx >= total) return;

    int d = idx % D;
    int t = idx / D;
    int h = t % HK; t /= HK;
    int s = t % S;
    int b = t / S;

    int pos = start[b] + s;
    size_t coff = (((size_t)b * L + pos) * HK + h) * D + d;
    size_t noff = (((size_t)(b * S + s)) * HK + h) * D + d;
    kc[coff] = kn[noff];
    vc[coff] = vn[noff];
}

// ---------------------------------------------------------------------------
// Flash attention per (b, hk). 2 waves; wave w owns query rows [16w,16w+16)
// of the 32 = G*S rows (row m -> s = m>>2, g = m&3).
// K/V tiles (16 keys x 128) are DMA'd into double-buffered LDS by the TDM
// (issued by wave 0, TENSORcnt-synchronized) so tile t+1 streams in while
// tile t is computed. Scores & PV via f32 WMMA; online softmax with shfl_xor
// reductions across the 16-lane N groups.
// ---------------------------------------------------------------------------
__global__ __launch_bounds__(64)
void attn_kernel(const float* __restrict__ q,
                 const float* __restrict__ kc,
                 const float* __restrict__ vc,
                 const int* __restrict__ ctx_len,
                 float* __restrict__ attn)   // (B,S,HQ,D)
{
    __shared__ __align__(16) float qbuf[32][D];
    __shared__ __align__(16) float kbuf[2][16][D];
    __shared__ __align__(16) float vbuf[2][16][D];
    __shared__ float pbuf[2][16][17];

    const int b    = blockIdx.x;
    const int hk   = blockIdx.y;
    const int tid  = threadIdx.x;
    const int wave = tid >> 5;
    const int lane = tid & 31;
    const int hl   = lane >> 4;
    const int r    = lane & 15;

    // --- load Q tile (32 rows x 128) cooperatively (one-time) ---
    const int RF4 = D / 4;
    for (int i = tid; i < 32 * RF4; i += 64) {
        int row = i / RF4, c4 = i % RF4;
        int s = row >> 2, g = row & 3;
        const v4f* src =
            (const v4f*)(q + (((size_t)(b * S + s) * HQ + hk * G + g) * D)) + c4;
        *((v4f*)&qbuf[row][c4 * 4]) = *src;
    }
    __syncthreads();

    // --- per-wave Q fragments (row = wave*16 + r, halves split K) ---
    v2f aq[32];
    {
        const int row = wave * 16 + r;
#pragma unroll
        for (int i = 0; i < 32; ++i)
            aq[i] = *(const v2f*)&qbuf[row][i * 4 + 2 * hl];
    }

    v8f o[8] = {};
    float run_m[8], run_l[8];
#pragma unroll
    for (int i = 0; i < 8; ++i) { run_m[i] = -1e30f; run_l[i] = 0.0f; }

    const int   ctx   = ctx_len[b];
    const float scale = 0.08838834764831845f;    // 1/sqrt(128)

    const size_t rstride = (size_t)HK * D;
    const float* kbase = kc + ((size_t)b * L * HK + hk) * D;
    const float* vbase = vc + ((size_t)b * L * HK + hk) * D;

    const int ntiles = ((ctx + S - 1) >> 4) + 1; // keys beyond ctx+S-1 masked

    const unsigned koff0 = (unsigned)(unsigned long long)&kbuf[0][0][0];
    const unsigned koff1 = (unsigned)(unsigned long long)&kbuf[1][0][0];
    const unsigned voff0 = (unsigned)(unsigned long long)&vbuf[0][0][0];
    const unsigned voff1 = (unsigned)(unsigned long long)&vbuf[1][0][0];

    // prologue: DMA tile 0 into buffer 0
    if (wave == 0) {
        tdm_load_tile_16x128(koff0, kbase);
        tdm_load_tile_16x128(voff0, vbase);
    }

    for (int t = 0; t < ntiles; ++t) {
        const int kt0 = t * 16;
        const int cur = t & 1;

        __syncthreads();   // all waves done reading the buffer we DMA into next
        if (wave == 0) {
            if (t + 1 < ntiles) {
                const int nxt = (t + 1) & 1;
                const float* kn = kbase + (size_t)(t + 1) * 16 * rstride;
                const float* vn = vbase + (size_t)(t + 1) * 16 * rstride;
                tdm_load_tile_16x128(nxt ? koff1 : koff0, kn);
                tdm_load_tile_16x128(nxt ? voff1 : voff0, vn);
                __builtin_amdgcn_s_wait_tensorcnt(2);  // tile t done; t+1 in flight
            } else {
                __builtin_amdgcn_s_wait_tensorcnt(0);  // last tile fully landed
            }
        }
        __syncthreads();   // tile t visible to both waves

        // scores: Qtile(16x128) @ Ktile^T(128x16)
        v8f sc = {};
#pragma unroll
        for (int i = 0; i < 32; ++i) {
            v2f bk = *(const v2f*)&kbuf[cur][r][i * 4 + 2 * hl];
            sc = __builtin_amdgcn_wmma_f32_16x16x4_f32(
                false, aq[i], false, bk, (short)0, sc, false, false);
        }

        // mask + scale + online softmax (row = wave*16 + hl*8 + i, key = kt0+r)
        float corr[8];
#pragma unroll
        for (int i = 0; i < 8; ++i) {
            int m = wave * 16 + hl * 8 + i;
            int s = m >> 2;
            float x = ((kt0 + r) <= (ctx + s)) ? sc[i] * scale : -1e30f;
            float rm = x;
            rm = fmaxf(rm, __shfl_xor(rm, 1));
            rm = fmaxf(rm, __shfl_xor(rm, 2));
            rm = fmaxf(rm, __shfl_xor(rm, 4));
            rm = fmaxf(rm, __shfl_xor(rm, 8));
            float nm = fmaxf(run_m[i], rm);
            float p  = __expf(x - nm);
            float rs = p;
            rs += __shfl_xor(rs, 1);
            rs += __shfl_xor(rs, 2);
            rs += __shfl_xor(rs, 4);
            rs += __shfl_xor(rs, 8);
            corr[i]  = __expf(run_m[i] - nm);
            run_l[i] = run_l[i] * corr[i] + rs;
            run_m[i] = nm;
            pbuf[wave][hl * 8 + i][r] = p;   // stash P for re-fragmenting
        }

        // rescale running output
#pragma unroll
        for (int tt = 0; tt < 8; ++tt)
#pragma unroll
            for (int i = 0; i < 8; ++i)
                o[tt][i] *= corr[i];

        // PV: P(16x16) @ Vtile(16x128); contract keys 4 at a time
#pragma unroll
        for (int kk = 0; kk < 4; ++kk) {
            v2f ap = { pbuf[wave][r][kk * 4 + 2 * hl],
                       pbuf[wave][r][kk * 4 + 2 * hl + 1] };
#pragma unroll
            for (int tt = 0; tt < 8; ++tt) {
                v2f bv = { vbuf[cur][kk * 4 + 2 * hl][tt * 16 + r],
                           vbuf[cur][kk * 4 + 2 * hl + 1][tt * 16 + r] };
                o[tt] = __builtin_amdgcn_wmma_f32_16x16x4_f32(
                    false, ap, false, bv, (short)0, o[tt], false, false);
            }
        }
    }

    // normalize and write attn (B,S,HQ,D)
#pragma unroll
    for (int i = 0; i < 8; ++i) {
        int m = wave * 16 + hl * 8 + i;
        int s = m >> 2, g = m & 3;
        float inv = 1.0f / run_l[i];
        size_t base = (((size_t)(b * S + s) * HQ) + hk * G + g) * D;
#pragma unroll
        for (int tt = 0; tt < 8; ++tt)
            attn[base + tt * 16 + r] = o[tt][i] * inv;
    }
}

// ---------------------------------------------------------------------------
extern "C" void kernel_launch(void* const* d_in, const int* in_sizes, int n_in,
                              void* d_out, int out_size, void* d_ws, size_t ws_size,
                              hipStream_t stream)
{
    const float* hs    = (const float*)d_in[0];  // (B,S,H)
    const float* pkv   = (const float*)d_in[1];  // (2,B,L,HK,D)
    const float* cs    = (const float*)d_in[2];  // (B,S,2,HALF)
    const int*   ctx   = (const int*)  d_in[3];  // (B,)
    const int*   start = (const int*)  d_in[4];  // (B,)
    const float* Wq    = (const float*)d_in[5];  // (HQ*D, H)
    const float* Wk    = (const float*)d_in[6];  // (HK*D, H)
    const float* Wv    = (const float*)d_in[7];  // (HK*D, H)
    const float* Wo    = (const float*)d_in[8];  // (H, HQ*D)

    float* out_attn = (float*)d_out;                         // (B,S,H)
    float* kcache   = out_attn + (size_t)B * S * H;          // (B,L,HK,D)
    float* vcache   = kcache + (size_t)B * L * HK * D;       // (B,L,HK,D)

    float* qb = (float*)d_ws;                                // (B,S,HQ,D)
    float* kb = qb + (size_t)B * S * HQ * D;                 // (B,S,HK,D)
    float* vb = kb + (size_t)B * S * HK * D;                 // (B,S,HK,D)
    float* ab = vb + (size_t)B * S * HK * D;                 // (B,S,HQ,D)

    const int M = B * S;  // 256

    // 1) QKV projections (WMMA f32)
    gemm_xwt_f32<<<dim3((HQ * D) / 64, M / 16), 128, 0, stream>>>(hs, Wq, qb, M, HQ * D, H);
    gemm_xwt_f32<<<dim3((HK * D) / 64, M / 16), 128, 0, stream>>>(hs, Wk, kb, M, HK * D, H);
    gemm_xwt_f32<<<dim3((HK * D) / 64, M / 16), 128, 0, stream>>>(hs, Wv, vb, M, HK * D, H);

    // 2) RoPE on q and k
    {
        int total = B * S * (HQ + HK) * HALF;
        rope_kernel<<<(total + 255) / 256, 256, 0, stream>>>(qb, kb, cs);
    }

    // 3) KV cache: bulk NT-load copy then scatter the S new rows
    {
        size_t n4 = (size_t)2 * B * L * HK * D / 4;
        cache_copy<<<8192, 256, 0, stream>>>((const v4f*)pkv, (v4f*)kcache, n4);
        int total = B * S * HK * D;
        cache_scatter<<<(total + 255) / 256, 256, 0, stream>>>(kb, vb, start, kcache, vcache);
    }

    // 4) Flash attention (WMMA f32 + TDM double-buffered K/V tiles)
    attn_kernel<<<dim3(B, HK), 64, 0, stream>>>(qb, kcache, vcache, ctx, ab);

    // 5) Output projection (WMMA f32)
    gemm_xwt_f32<<<dim3(H / 64, M / 16), 128, 0, stream>>>(ab, Wo, out_attn, M, H, HQ * D);
}